// teamGNN_14731737825584
// MI455X (gfx1250) — compile-verified
//
#include <hip/hip_runtime.h>

#define N_NODES 100000
#define N_EDGES 1600000
#define D_FEAT  128
#define HIDDEN  64

typedef float v2f __attribute__((ext_vector_type(2)));
typedef float v8f __attribute__((ext_vector_type(8)));

// HW fp32 atomic add (global_atomic_add_f32), bypassing CAS-loop lowering.
__device__ __forceinline__ void atomAddF32(float* addr, float v) {
#if defined(__HIP_DEVICE_COMPILE__)
    unsafeAtomicAdd(addr, v);
#else
    atomicAdd(addr, v);
#endif
}

// ---------------- degree / norm precompute ----------------

__global__ void k_zero(float* __restrict__ p, int n) {
    int i = blockIdx.x * blockDim.x + threadIdx.x;
    if (i < n) p[i] = 0.0f;
}

__global__ void k_degree(const int* __restrict__ col, const float* __restrict__ ew,
                         float* __restrict__ deg, int E) {
    int e = blockIdx.x * blockDim.x + threadIdx.x;
    if (e < E) atomAddF32(&deg[col[e]], ew[e]);
}

__global__ void k_dinv(const float* __restrict__ deg, float* __restrict__ dinv, int n) {
    int i = blockIdx.x * blockDim.x + threadIdx.x;
    if (i < n) dinv[i] = rsqrtf(deg[i] + 1.0f);   // self-loop weight 1 => deg_total >= 1
}

__global__ void k_norm(const int* __restrict__ row, const int* __restrict__ col,
                       const float* __restrict__ ew, const float* __restrict__ dinv,
                       float* __restrict__ nrm, int E) {
    int e = blockIdx.x * blockDim.x + threadIdx.x;
    if (e < E) nrm[e] = dinv[row[e]] * ew[e] * dinv[col[e]];
}

// ---------------- GEMM1: XW = X @ W1 via V_WMMA_F32_16X16X4_F32 ----------------
// One wave computes a 16x64 tile (4 x 16x16 accumulators). 8 waves/block = 128 rows.
// W1 staged in LDS k-pair interleaved: sW[(k/2)*128 + 2n + (k&1)] = W1[k][n],
// so a B fragment {W[k][n], W[k+1][n]} is a single aligned ds_load_b64 into a
// consecutive VGPR pair (no register shuffling between WMMAs).

__global__ __launch_bounds__(256)
void k_gemm1_wmma(const float* __restrict__ X, const float* __restrict__ W1,
                  float* __restrict__ XW) {
    __shared__ float sW[D_FEAT * HIDDEN];   // 32 KB, pair-interleaved
    int tid = threadIdx.x;
    #pragma unroll
    for (int i = tid; i < D_FEAT * HIDDEN; i += 256) {
        int k = i >> 6;          // row of W1
        int n = i & 63;          // col of W1
        sW[(k >> 1) * 128 + (n << 1) + (k & 1)] = W1[i];
    }
    __syncthreads();

    const int wave = tid >> 5;
    const int lane = tid & 31;
    const int half = lane >> 4;     // 0: lanes 0-15, 1: lanes 16-31
    const int l16  = lane & 15;
    const int rowbase = blockIdx.x * 128 + wave * 16;

    // A-fragment row for this lane (both halves cover M = 0..15)
    int m = rowbase + l16;
    if (m >= N_NODES) m = N_NODES - 1;          // clamp; garbage rows are never stored
    const float* xrow = X + (size_t)m * D_FEAT + 2 * half;

    v8f acc0 = {}, acc1 = {}, acc2 = {}, acc3 = {};

    #pragma unroll
    for (int k = 0; k < D_FEAT; k += 4) {
        // A 16x4 fp32 fragment: VGPR0 = K (lanes 0-15) / K+2 (lanes 16-31); VGPR1 = +1
        v2f a = *reinterpret_cast<const v2f*>(xrow + k);    // 8B aligned

        // B pair rows: lanes 0-15 get rows (k,k+1); lanes 16-31 get rows (k+2,k+3)
        const v2f* brow = reinterpret_cast<const v2f*>(sW + ((k >> 1) + half) * 128) + l16;
        acc0 = __builtin_amdgcn_wmma_f32_16x16x4_f32(false, a, false, brow[0],  (short)0, acc0, false, false);
        acc1 = __builtin_amdgcn_wmma_f32_16x16x4_f32(false, a, false, brow[16], (short)0, acc1, false, false);
        acc2 = __builtin_amdgcn_wmma_f32_16x16x4_f32(false, a, false, brow[32], (short)0, acc2, false, false);
        acc3 = __builtin_amdgcn_wmma_f32_16x16x4_f32(false, a, false, brow[48], (short)0, acc3, false, false);
    }

    // C/D layout: VGPR v holds (M = v + 8*half, N = l16)
    #pragma unroll
    for (int v = 0; v < 8; ++v) {
        int mm = rowbase + v + 8 * half;
        if (mm < N_NODES) {
            float* dst = XW + (size_t)mm * HIDDEN + l16;
            dst[0]  = acc0[v];
            dst[16] = acc1[v];
            dst[32] = acc2[v];
            dst[48] = acc3[v];
        }
    }
}

// ---------------- layer-1 aggregation ----------------

// Self-loop term, full overwrite (no zero-init needed): agg = XW * dinv^2
__global__ void k_selfinit(const float* __restrict__ XW, const float* __restrict__ dinv,
                           float* __restrict__ agg, int total) {
    int i = blockIdx.x * blockDim.x + threadIdx.x;
    if (i < total) {
        float dv = dinv[i >> 6];
        agg[i] = XW[i] * dv * dv;
    }
}

// 16 threads per edge; each handles 4 contiguous cols via float4 gather + 4 HW atomics.
__global__ __launch_bounds__(256)
void k_edgeagg1(const int* __restrict__ row, const int* __restrict__ col,
                const float* __restrict__ nrm, const float* __restrict__ XW,
                float* __restrict__ agg) {
    int gid = blockIdx.x * blockDim.x + threadIdx.x;
    int e = gid >> 4;
    if (e >= N_EDGES) return;
    int c4 = gid & 15;
    int r = row[e];
    int t = col[e];
    float w = nrm[e];
    float4 v = reinterpret_cast<const float4*>(XW + (size_t)r * HIDDEN)[c4];
    float* dst = agg + (size_t)t * HIDDEN + (c4 << 2);
    atomAddF32(dst + 0, v.x * w);
    atomAddF32(dst + 1, v.y * w);
    atomAddF32(dst + 2, v.z * w);
    atomAddF32(dst + 3, v.w * w);
}

// ---------------- layer 2: z = relu(agg + b1) @ W2, wave32 reduction ----------------

__global__ __launch_bounds__(256)
void k_layer2_dot(const float* __restrict__ agg, const float* __restrict__ b1,
                  const float* __restrict__ W2, float* __restrict__ z) {
    int lane = threadIdx.x & 31;
    int node = blockIdx.x * 8 + (threadIdx.x >> 5);
    if (node >= N_NODES) return;
    const float* a = agg + (size_t)node * HIDDEN;
    float h0 = a[lane]      + b1[lane];      h0 = h0 > 0.0f ? h0 : 0.0f;
    float h1 = a[lane + 32] + b1[lane + 32]; h1 = h1 > 0.0f ? h1 : 0.0f;
    float p = h0 * W2[lane] + h1 * W2[lane + 32];
    #pragma unroll
    for (int off = 16; off > 0; off >>= 1) p += __shfl_xor(p, off, 32);
    if (lane == 0) z[node] = p;
}

// out = z*dinv^2 + b2 (self-loop + bias, full overwrite)
__global__ void k_outinit(const float* __restrict__ z, const float* __restrict__ dinv,
                          const float* __restrict__ b2, float* __restrict__ out, int n) {
    int i = blockIdx.x * blockDim.x + threadIdx.x;
    if (i < n) {
        float dv = dinv[i];
        out[i] = z[i] * dv * dv + b2[0];
    }
}

__global__ void k_edgeagg2(const int* __restrict__ row, const int* __restrict__ col,
                           const float* __restrict__ nrm, const float* __restrict__ z,
                           float* __restrict__ out, int E) {
    int e = blockIdx.x * blockDim.x + threadIdx.x;
    if (e < E) atomAddF32(&out[col[e]], z[row[e]] * nrm[e]);
}

// ---------------- launch ----------------

extern "C" void kernel_launch(void* const* d_in, const int* in_sizes, int n_in,
                              void* d_out, int out_size, void* d_ws, size_t ws_size,
                              hipStream_t stream) {
    const float* x   = (const float*)d_in[0];
    const int*   ei  = (const int*)  d_in[1];
    const float* ew  = (const float*)d_in[2];
    const float* W1  = (const float*)d_in[3];
    const float* b1  = (const float*)d_in[4];
    const float* W2  = (const float*)d_in[5];
    const float* b2  = (const float*)d_in[6];
    float* out = (float*)d_out;

    const int* row = ei;            // edge_index[0] = source
    const int* col = ei + N_EDGES;  // edge_index[1] = target

    float* ws   = (float*)d_ws;
    float* deg  = ws;                          // N
    float* dinv = deg  + N_NODES;              // N
    float* nrm  = dinv + N_NODES;              // E
    float* XW   = nrm  + N_EDGES;              // N*64
    float* agg  = XW   + (size_t)N_NODES * 64; // N*64
    float* z    = agg  + (size_t)N_NODES * 64; // N

    const int B = 256;
    // degree + normalization
    k_zero   <<<(N_NODES + B - 1) / B, B, 0, stream>>>(deg, N_NODES);
    k_degree <<<(N_EDGES + B - 1) / B, B, 0, stream>>>(col, ew, deg, N_EDGES);
    k_dinv   <<<(N_NODES + B - 1) / B, B, 0, stream>>>(deg, dinv, N_NODES);
    k_norm   <<<(N_EDGES + B - 1) / B, B, 0, stream>>>(row, col, ew, dinv, nrm, N_EDGES);

    // layer 1
    k_gemm1_wmma <<<(N_NODES + 127) / 128, B, 0, stream>>>(x, W1, XW);
    k_selfinit   <<<((N_NODES * 64) + B - 1) / B, B, 0, stream>>>(XW, dinv, agg, N_NODES * 64);
    k_edgeagg1   <<<(N_EDGES * 16) / B, B, 0, stream>>>(row, col, nrm, XW, agg);

    // layer 2
    k_layer2_dot <<<(N_NODES + 7) / 8, B, 0, stream>>>(agg, b1, W2, z);
    k_outinit    <<<(N_NODES + B - 1) / B, B, 0, stream>>>(z, dinv, b2, out, N_NODES);
    k_edgeagg2   <<<(N_EDGES + B - 1) / B, B, 0, stream>>>(row, col, nrm, z, out, N_EDGES);
}